// GCN2_43937515438591
// MI455X (gfx1250) — compile-verified
//
#include <hip/hip_runtime.h>
#include <math.h>

// GCN2 (GCNII) forward: N=100000 nodes, E=1.6M edges, F_IN=512, H=128, C=40, L=8
#define N_NODES 100000
#define NE      1600000
#define F_IN    512
#define H       128
#define NC      40
#define L_LAYERS 8
#define ALPHA   0.1f
#define THETA   0.5f
#define CT      3   // ceil(48/16) column tiles for the C=40 output GEMM

typedef float v2f __attribute__((ext_vector_type(2)));
typedef float v8f __attribute__((ext_vector_type(8)));

// One K=4 step of D = A(16x4,f32) * B(4x16,f32) + C(16x16,f32), full f32 precision.
__device__ __forceinline__ v8f wmma_f32_k4(v2f a, v2f b, v8f c) {
  return __builtin_amdgcn_wmma_f32_16x16x4_f32(
      /*neg_a=*/false, a, /*neg_b=*/false, b,
      /*c_mod=*/(short)0, c, /*reuse_a=*/false, /*reuse_b=*/false);
}

// ---------------------------------------------------------------------------
// Kernel 1: h = relu(x @ W0 + b0); also writes x0 = h (initial residual).
// One 16x16 output tile per wave; 8 waves (256 thr) per block. K=512 -> 128 wmma.
// ---------------------------------------------------------------------------
__global__ __launch_bounds__(256)
void gemm_in(const float* __restrict__ x, const float* __restrict__ W0,
             const float* __restrict__ b0,
             float* __restrict__ h, float* __restrict__ x0) {
  const int lane  = threadIdx.x & 31;
  const int wave  = blockIdx.x * 8 + (threadIdx.x >> 5);
  const int tr    = wave >> 3;          // node-row tile (H/16 = 8 col tiles)
  const int tc    = wave & 7;
  const int row   = tr * 16 + (lane & 15);   // A fragment: M = lane%16
  const int col   = tc * 16 + (lane & 15);   // B fragment: N = lane%16
  const int kh    = (lane >> 4) << 1;        // lane half selects K pair {0,1} / {2,3}

  const float* arow = x + row * F_IN;
  v8f acc = {};
#pragma unroll 4
  for (int k = 0; k < F_IN; k += 4) {
    v2f a, b;
    a.x = arow[k + kh];
    a.y = arow[k + kh + 1];
    b.x = W0[(k + kh)     * H + col];
    b.y = W0[(k + kh + 1) * H + col];
    acc = wmma_f32_k4(a, b, acc);
  }
  const int  r0   = tr * 16 + ((lane >> 4) << 3);  // C/D: M = i + 8*(lane>=16)
  const float bias = b0[col];
#pragma unroll
  for (int i = 0; i < 8; ++i) {
    const float v = fmaxf(acc[i] + bias, 0.0f);
    const int idx = (r0 + i) * H + col;
    h[idx]  = v;
    x0[idx] = v;
  }
}

// ---------------------------------------------------------------------------
// Kernel 2: edge scatter  agg[dst] += edge_weight * h[src]
// One wave per edge; lane covers features {lane, lane+32, lane+64, lane+96}.
// Coalesced 128B loads + coalesced f32 atomics; L2-resident (h,agg = 51MB each).
// ---------------------------------------------------------------------------
__global__ __launch_bounds__(256)
void scatter_edges(const int* __restrict__ src, const int* __restrict__ dst,
                   const float* __restrict__ ew, const float* __restrict__ h,
                   float* __restrict__ agg) {
  const int lane = threadIdx.x & 31;
  const int e    = blockIdx.x * 8 + (threadIdx.x >> 5);   // NE = 200000*8 exactly
  const int s    = src[e];
  const int d    = dst[e];
  const float w  = ew[e];
#pragma unroll
  for (int j = 0; j < 4; ++j) {
    const int f = lane + 32 * j;
    atomicAdd(&agg[d * H + f], h[s * H + f] * w);
  }
}

// ---------------------------------------------------------------------------
// Kernel 3: fused GCN2 layer update.
//   out = (1-ALPHA)*agg + ALPHA*x0            (computed on the fly)
//   h   = relu((1-beta)*out + beta*(out @ W))
// K = H = 128 -> 32 wmma per tile. Reads agg/x0, writes h (safe to overwrite).
// ---------------------------------------------------------------------------
__global__ __launch_bounds__(256)
void gemm_layer(const float* __restrict__ agg, const float* __restrict__ x0,
                const float* __restrict__ W, float* __restrict__ hout, float beta) {
  const int lane = threadIdx.x & 31;
  const int wave = blockIdx.x * 8 + (threadIdx.x >> 5);
  const int tr   = wave >> 3;
  const int tc   = wave & 7;
  const int row  = tr * 16 + (lane & 15);
  const int col  = tc * 16 + (lane & 15);
  const int kh   = (lane >> 4) << 1;
  const float oma = 1.0f - ALPHA;

  v8f acc = {};
#pragma unroll 4
  for (int k = 0; k < H; k += 4) {
    const int ka = k + kh;
    v2f a, b;
    a.x = oma * agg[row * H + ka]     + ALPHA * x0[row * H + ka];
    a.y = oma * agg[row * H + ka + 1] + ALPHA * x0[row * H + ka + 1];
    b.x = W[ka * H + col];
    b.y = W[(ka + 1) * H + col];
    acc = wmma_f32_k4(a, b, acc);
  }
  const int   r0  = tr * 16 + ((lane >> 4) << 3);
  const float omb = 1.0f - beta;
#pragma unroll
  for (int i = 0; i < 8; ++i) {
    const int   idx  = (r0 + i) * H + col;
    const float outv = oma * agg[idx] + ALPHA * x0[idx];
    hout[idx] = fmaxf(omb * outv + beta * acc[i], 0.0f);
  }
}

// ---------------------------------------------------------------------------
// Kernel 4: logits = h @ W1 + b1  ->  d_out (raw, before softmax).
// C=40 padded to 48 (3 col tiles); B columns >= 40 read as 0, stores guarded.
// Guard is before any wmma, wave-uniform -> EXEC all-ones at every matrix op.
// ---------------------------------------------------------------------------
__global__ __launch_bounds__(256)
void gemm_out(const float* __restrict__ h, const float* __restrict__ W1,
              const float* __restrict__ b1, float* __restrict__ logits) {
  const int lane = threadIdx.x & 31;
  const int wave = blockIdx.x * 8 + (threadIdx.x >> 5);
  if (wave >= (N_NODES / 16) * CT) return;   // wave-uniform guard
  const int tr   = wave / CT;
  const int tc   = wave % CT;
  const int row  = tr * 16 + (lane & 15);
  const int col  = tc * 16 + (lane & 15);
  const int kh   = (lane >> 4) << 1;
  const bool cok = (col < NC);

  v8f acc = {};
#pragma unroll 4
  for (int k = 0; k < H; k += 4) {
    const int ka = k + kh;
    v2f a, b;
    a.x = h[row * H + ka];
    a.y = h[row * H + ka + 1];
    b.x = cok ? W1[ka * NC + col]       : 0.0f;   // v_cndmask, EXEC untouched
    b.y = cok ? W1[(ka + 1) * NC + col] : 0.0f;
    acc = wmma_f32_k4(a, b, acc);
  }
  if (cok) {
    const int   r0   = tr * 16 + ((lane >> 4) << 3);
    const float bias = b1[col];
#pragma unroll
    for (int i = 0; i < 8; ++i)
      logits[(r0 + i) * NC + col] = acc[i] + bias;
  }
}

// ---------------------------------------------------------------------------
// Kernel 5: in-place row-wise log_softmax over d_out. One wave32 per row;
// lane covers cols {lane, lane+32}; shuffle max/sum reductions (exact expf/logf).
// ---------------------------------------------------------------------------
__global__ __launch_bounds__(256)
void log_softmax_rows(float* __restrict__ logits) {
  const int lane = threadIdx.x & 31;
  const int row  = blockIdx.x * 8 + (threadIdx.x >> 5);   // N = 12500*8 exactly
  const bool ok1 = (lane + 32) < NC;
  const float v0 = logits[row * NC + lane];               // lane < 32 <= NC always valid
  const float v1 = ok1 ? logits[row * NC + lane + 32] : -INFINITY;

  float m = fmaxf(v0, v1);
#pragma unroll
  for (int off = 16; off > 0; off >>= 1)
    m = fmaxf(m, __shfl_xor(m, off, 32));

  float s = expf(v0 - m) + (ok1 ? expf(v1 - m) : 0.0f);
#pragma unroll
  for (int off = 16; off > 0; off >>= 1)
    s += __shfl_xor(s, off, 32);

  const float ls = logf(s) + m;
  logits[row * NC + lane] = v0 - ls;
  if (ok1) logits[row * NC + lane + 32] = v1 - ls;
}

// ---------------------------------------------------------------------------
extern "C" void kernel_launch(void* const* d_in, const int* in_sizes, int n_in,
                              void* d_out, int out_size, void* d_ws, size_t ws_size,
                              hipStream_t stream) {
  (void)in_sizes; (void)n_in; (void)out_size; (void)ws_size;
  const float* x     = (const float*)d_in[0];
  const int*   ei    = (const int*)  d_in[1];   // [2, E]
  const float* ew    = (const float*)d_in[2];
  const float* W0    = (const float*)d_in[3];
  const float* b0    = (const float*)d_in[4];
  const float* convW = (const float*)d_in[5];   // [L, H, H]
  const float* W1    = (const float*)d_in[6];
  const float* b1    = (const float*)d_in[7];
  float*       out   = (float*)d_out;

  const int* src = ei;
  const int* dst = ei + NE;

  // Workspace: h, x0, agg — 51.2 MB each, all L2-resident together (192 MB L2).
  float* h   = (float*)d_ws;
  float* x0  = h  + (size_t)N_NODES * H;
  float* agg = x0 + (size_t)N_NODES * H;

  const dim3 blk(256);
  const int  gemm_blocks = (N_NODES / 16) * (H / 16) / 8;   // 6250

  gemm_in<<<gemm_blocks, blk, 0, stream>>>(x, W0, b0, h, x0);

  for (int l = 0; l < L_LAYERS; ++l) {
    const float beta = logf(THETA / (float)(l + 1) + 1.0f);
    hipMemsetAsync(agg, 0, (size_t)N_NODES * H * sizeof(float), stream);
    scatter_edges<<<NE / 8, blk, 0, stream>>>(src, dst, ew, h, agg);
    gemm_layer<<<gemm_blocks, blk, 0, stream>>>(agg, x0, convW + (size_t)l * H * H, h, beta);
  }

  const int waves3 = (N_NODES / 16) * CT;                   // 18750
  gemm_out<<<(waves3 + 7) / 8, blk, 0, stream>>>(h, W1, b1, out);
  log_softmax_rows<<<N_NODES / 8, blk, 0, stream>>>(out);
}